// BBoxHead_11673721111147
// MI455X (gfx1250) — compile-verified
//
#include <hip/hip_runtime.h>
#include <math.h>

typedef __attribute__((ext_vector_type(2))) float v2f;
typedef __attribute__((ext_vector_type(4))) float v4f;
typedef __attribute__((ext_vector_type(8))) float v8f;

#define TM 128
#define TN 64
#define TK 16
#define LDPAD 20   // padded row stride (floats): 80B, 16B-aligned, bank-conflict-free
#define BN_EPS 1e-3f

// ---------------------------------------------------------------------------
// Generic f32 WMMA GEMM with optional fused BN+ReLU on the A operand:
//   C[M,N] = act(A)[M,K] * B[K,N] + bias[N]
//   act(a)[m,k] = aScale ? relu(a[m,k]*aScale[k] + aShift[k]) : a[m,k]
// Block: 256 threads (8 waves). Block tile 128x64, K-chunk 16.
// Wave (wm 0..3, wn 0..1) computes a 32x32 tile = 4 v_wmma_f32_16x16x4_f32 accums.
// Interior blocks: software-pipelined (regs prefetch chunk kt+1 during compute
// of kt) + double-buffered LDS -> one barrier per K-chunk.
// Edge blocks: simple guarded single-buffer path.
// ---------------------------------------------------------------------------
__global__ __launch_bounds__(256)
void gemm_f32_wmma(const float* __restrict__ A, const float* __restrict__ B,
                   const float* __restrict__ bias,
                   const float* __restrict__ aScale, const float* __restrict__ aShift,
                   float* __restrict__ C, int M, int N, int K) {
  __shared__ float Alds[2][TM * LDPAD];   // A tiles, [m][k], row-padded
  __shared__ float Blds[2][TN * LDPAD];   // B tiles TRANSPOSED: [n][k], row-padded

  const int tid  = threadIdx.x;
  const int lane = tid & 31;
  const int wave = tid >> 5;
  const int wm   = wave & 3;    // M sub-tile (32 rows each)
  const int wn   = wave >> 2;   // N sub-tile (32 cols each)
  const int l16  = lane & 15;
  const int half = lane >> 4;   // 0: K=0,1 ; 1: K=2,3  (ISA 16x4 / 4x16 layout)

  const int m0 = blockIdx.y * TM;
  const int n0 = blockIdx.x * TN;

  v8f c00 = {}; v8f c01 = {}; v8f c10 = {}; v8f c11 = {};

  // A global->LDS mapping: each thread moves two float4 (rows tid/4 and tid/4+64)
  const int arow = tid >> 2;           // 0..63
  const int aseg = tid & 3;            // float4 segment within the 16-wide K chunk
  // B global->LDS mapping: each thread moves one float4 (row k=tid/16, 4 cols)
  const int bk   = tid >> 4;           // 0..15  (k within chunk)
  const int bn4  = (tid & 15) << 2;    // 0,4,...,60 (n within tile)

  const int r0 = m0 + arow;
  const int r1 = m0 + arow + 64;
  const int nk = K / TK;               // K is always a multiple of 16 here
  const bool edge = (m0 + TM > M) || (n0 + TN > N);

  // WMMA compute on one LDS buffer (16 x v_wmma_f32_16x16x4_f32).
  auto compute = [&](int buf) {
    const float* Ab0 = &Alds[buf][(wm * 32 + l16) * LDPAD + half * 2];
    const float* Ab1 = Ab0 + 16 * LDPAD;
    const float* Bb0 = &Blds[buf][(wn * 32 + l16) * LDPAD + half * 2];
    const float* Bb1 = Bb0 + 16 * LDPAD;
#pragma unroll
    for (int kk = 0; kk < TK; kk += 4) {
      const v2f fa0 = *(const v2f*)(Ab0 + kk);
      const v2f fa1 = *(const v2f*)(Ab1 + kk);
      const v2f fb0 = *(const v2f*)(Bb0 + kk);
      const v2f fb1 = *(const v2f*)(Bb1 + kk);
      c00 = __builtin_amdgcn_wmma_f32_16x16x4_f32(false, fa0, false, fb0, (short)0, c00, false, false);
      c01 = __builtin_amdgcn_wmma_f32_16x16x4_f32(false, fa0, false, fb1, (short)0, c01, false, false);
      c10 = __builtin_amdgcn_wmma_f32_16x16x4_f32(false, fa1, false, fb0, (short)0, c10, false, false);
      c11 = __builtin_amdgcn_wmma_f32_16x16x4_f32(false, fa1, false, fb1, (short)0, c11, false, false);
    }
  };

  // Stage a chunk (held in VGPRs) into LDS buffer `buf`.
  auto stage = [&](int buf, v4f a0, v4f a1, v4f b4) {
    *(v4f*)&Alds[buf][arow * LDPAD + aseg * 4]        = a0;
    *(v4f*)&Alds[buf][(arow + 64) * LDPAD + aseg * 4] = a1;
#pragma unroll
    for (int j = 0; j < 4; ++j) Blds[buf][(bn4 + j) * LDPAD + bk] = b4[j];
  };

  // Fused BN(train)+ReLU on staged A registers.
  auto fuse = [&](v4f& a0, v4f& a1, const float* s, const float* h) {
    const v4f s4 = *(const v4f*)s;
    const v4f h4 = *(const v4f*)h;
#pragma unroll
    for (int j = 0; j < 4; ++j) {
      a0[j] = fmaxf(fmaf(a0[j], s4[j], h4[j]), 0.0f);
      a1[j] = fmaxf(fmaf(a1[j], s4[j], h4[j]), 0.0f);
    }
  };

  if (!edge) {
    // ---------------- interior: pipelined + double-buffered ----------------
    const float* pA0 = A + (long)r0 * K + aseg * 4;
    const float* pA1 = A + (long)r1 * K + aseg * 4;
    const float* pB  = B + (long)bk * N + n0 + bn4;
    const float* pS  = aScale ? aScale + aseg * 4 : nullptr;
    const float* pH  = aScale ? aShift + aseg * 4 : nullptr;
    const long   bstep = (long)TK * N;

    {  // prologue: stage chunk 0 into buffer 0
      v4f a0 = *(const v4f*)pA0;
      v4f a1 = *(const v4f*)pA1;
      v4f b4 = *(const v4f*)pB;
      if (aScale) fuse(a0, a1, pS, pH);
      stage(0, a0, a1, b4);
    }
    __syncthreads();

    for (int kt = 0; kt < nk - 1; ++kt) {
      // issue global loads for chunk kt+1 (consumed only after compute)
      pA0 += TK; pA1 += TK; pB += bstep;
      v4f a0 = *(const v4f*)pA0;
      v4f a1 = *(const v4f*)pA1;
      v4f b4 = *(const v4f*)pB;
      __builtin_prefetch(pA0 + TK, 0, 1);          // speculative: chunk kt+2
      __builtin_prefetch(pA1 + TK, 0, 1);
      __builtin_prefetch(pB + bstep, 0, 1);

      compute(kt & 1);                              // 16 WMMAs hide load latency

      if (aScale) { pS += TK; pH += TK; fuse(a0, a1, pS, pH); }
      stage((kt + 1) & 1, a0, a1, b4);              // other buffer: no hazard
      __syncthreads();                              // one barrier per chunk
    }
    compute((nk - 1) & 1);
  } else {
    // --------------------------- edge (guarded) ---------------------------
    for (int kt = 0; kt < nk; ++kt) {
      const int kbase = kt * TK;
      v4f a0 = {}; v4f a1 = {};
      if (r0 < M) a0 = *(const v4f*)&A[(long)r0 * K + kbase + aseg * 4];
      if (r1 < M) a1 = *(const v4f*)&A[(long)r1 * K + kbase + aseg * 4];
      if (aScale) fuse(a0, a1, &aScale[kbase + aseg * 4], &aShift[kbase + aseg * 4]);
      float bel[4];
#pragma unroll
      for (int j = 0; j < 4; ++j) {
        const int col = n0 + bn4 + j;
        bel[j] = (col < N) ? B[(long)(kbase + bk) * N + col] : 0.0f;
      }
      __syncthreads();
      *(v4f*)&Alds[0][arow * LDPAD + aseg * 4]        = a0;
      *(v4f*)&Alds[0][(arow + 64) * LDPAD + aseg * 4] = a1;
#pragma unroll
      for (int j = 0; j < 4; ++j) Blds[0][(bn4 + j) * LDPAD + bk] = bel[j];
      __syncthreads();
      compute(0);
    }
  }

  // ---- epilogue: C/D layout — VGPR i: lanes0-15 -> M=i, lanes16-31 -> M=i+8 ----
#pragma unroll
  for (int ti = 0; ti < 2; ++ti) {
#pragma unroll
    for (int tj = 0; tj < 2; ++tj) {
      const v8f cc = (ti == 0) ? ((tj == 0) ? c00 : c01)
                               : ((tj == 0) ? c10 : c11);
      const int col   = n0 + wn * 32 + tj * 16 + l16;
      const int rbase = m0 + wm * 32 + ti * 16 + half * 8;
      if (col < N) {
        const float bv = bias ? bias[col] : 0.0f;
#pragma unroll
        for (int i = 0; i < 8; ++i) {
          const int r = rbase + i;
          if (r < M) C[(long)r * N + col] = cc[i] + bv;
        }
      }
    }
  }
}

// ---------------------------------------------------------------------------
// Batch-norm (training) stats per column: scale = gamma*rsqrt(var+eps),
// shift = beta - mean*scale.  var = E[x^2] - mean^2 (population, as jnp.var).
// One block handles 64 columns; threads are 64 cols x 4 row-groups (coalesced).
// ---------------------------------------------------------------------------
__global__ __launch_bounds__(256)
void bn_stats(const float* __restrict__ Y, const float* __restrict__ gamma,
              const float* __restrict__ beta, float* __restrict__ scale,
              float* __restrict__ shift, int M, int Ncols) {
  __shared__ float ls [4][64];
  __shared__ float ls2[4][64];
  const int c   = threadIdx.x & 63;
  const int rg  = threadIdx.x >> 6;      // 0..3
  const int col = blockIdx.x * 64 + c;
  float s = 0.0f, s2 = 0.0f;
  for (int r = rg; r < M; r += 4) {
    const float v = Y[(long)r * Ncols + col];
    s += v; s2 += v * v;
  }
  ls[rg][c] = s; ls2[rg][c] = s2;
  __syncthreads();
  if (rg == 0) {
    float t  = ls[0][c] + ls[1][c] + ls[2][c] + ls[3][c];
    float t2 = ls2[0][c] + ls2[1][c] + ls2[2][c] + ls2[3][c];
    const float inv_m = 1.0f / (float)M;
    const float mean = t * inv_m;
    const float var  = t2 * inv_m - mean * mean;
    const float sc   = gamma[col] * rsqrtf(var + BN_EPS);
    scale[col] = sc;
    shift[col] = beta[col] - mean * sc;
  }
}

// Softmax over Ncls (=81) per row; one wave32 per row, shuffle reductions.
__global__ __launch_bounds__(256)
void softmax_rows(const float* __restrict__ logits, float* __restrict__ probs,
                  int M, int Ncls) {
  const int wave = (int)((blockIdx.x * (long)blockDim.x + threadIdx.x) >> 5);
  const int lane = threadIdx.x & 31;
  if (wave >= M) return;
  const float* row = logits + (long)wave * Ncls;
  float*       out = probs  + (long)wave * Ncls;
  float mx = -3.4e38f;
  for (int j = lane; j < Ncls; j += 32) mx = fmaxf(mx, row[j]);
#pragma unroll
  for (int off = 16; off > 0; off >>= 1) mx = fmaxf(mx, __shfl_xor(mx, off, 32));
  float s = 0.0f;
  for (int j = lane; j < Ncls; j += 32) {
    const float e = __expf(row[j] - mx);
    out[j] = e;
    s += e;
  }
#pragma unroll
  for (int off = 16; off > 0; off >>= 1) s += __shfl_xor(s, off, 32);
  const float inv = 1.0f / s;
  for (int j = lane; j < Ncls; j += 32) out[j] *= inv;
}

// ---------------------------------------------------------------------------
extern "C" void kernel_launch(void* const* d_in, const int* in_sizes, int n_in,
                              void* d_out, int out_size, void* d_ws, size_t ws_size,
                              hipStream_t stream) {
  (void)in_sizes; (void)n_in; (void)out_size; (void)ws_size;
  const int N   = 2000;
  const int K1  = 7 * 7 * 256;   // 12544
  const int HID = 1024;
  const int NC  = 81;

  const float* rois     = (const float*)d_in[0];
  const float* w1       = (const float*)d_in[1];
  const float* b1       = (const float*)d_in[2];
  const float* gamma1   = (const float*)d_in[3];
  const float* beta1    = (const float*)d_in[4];
  const float* w2       = (const float*)d_in[5];
  const float* b2       = (const float*)d_in[6];
  const float* gamma2   = (const float*)d_in[7];
  const float* beta2    = (const float*)d_in[8];
  const float* w_logits = (const float*)d_in[9];
  const float* b_logits = (const float*)d_in[10];
  const float* w_delta  = (const float*)d_in[11];
  const float* b_delta  = (const float*)d_in[12];

  float* ws  = (float*)d_ws;
  float* Y1  = ws;                                  // [2000,1024] raw pre-activation of FC1
  float* Y2  = ws + (size_t)N * HID;                // [2000,1024] raw pre-activation of FC2
  float* sc1 = ws + (size_t)2 * N * HID;
  float* sh1 = sc1 + HID;
  float* sc2 = sh1 + HID;
  float* sh2 = sc2 + HID;

  float* logits = (float*)d_out;                    // [2000,81]
  float* probs  = logits + (size_t)N * NC;          // [2000,81]
  float* deltas = logits + (size_t)2 * N * NC;      // [2000,324]

  const dim3 blk(256);
  const int mblk = (N + TM - 1) / TM;               // 16

  // FC1 (7x7 VALID conv == flattened dense) -> raw Y1, then BN stats
  gemm_f32_wmma<<<dim3(HID / TN, mblk), blk, 0, stream>>>(
      rois, w1, b1, nullptr, nullptr, Y1, N, HID, K1);
  bn_stats<<<HID / 64, blk, 0, stream>>>(Y1, gamma1, beta1, sc1, sh1, N, HID);

  // FC2: BN1+ReLU fused into A-load; -> raw Y2, then BN stats
  gemm_f32_wmma<<<dim3(HID / TN, mblk), blk, 0, stream>>>(
      Y1, w2, b2, sc1, sh1, Y2, N, HID, HID);
  bn_stats<<<HID / 64, blk, 0, stream>>>(Y2, gamma2, beta2, sc2, sh2, N, HID);

  // Heads: BN2+ReLU fused into A-load, straight into d_out
  // (deltas [N,81,4] is layout-identical to [N,324])
  gemm_f32_wmma<<<dim3((NC + TN - 1) / TN, mblk), blk, 0, stream>>>(
      Y2, w_logits, b_logits, sc2, sh2, logits, N, NC, HID);
  gemm_f32_wmma<<<dim3((NC * 4 + TN - 1) / TN, mblk), blk, 0, stream>>>(
      Y2, w_delta, b_delta, sc2, sh2, deltas, N, NC * 4, HID);

  // Softmax: one wave per row
  softmax_rows<<<(N * 32 + 255) / 256, blk, 0, stream>>>(logits, probs, N, NC);
}